// Net_graph_46849503265417
// MI455X (gfx1250) — compile-verified
//
#include <hip/hip_runtime.h>
#include <hip/hip_bf16.h>

typedef __attribute__((ext_vector_type(2))) float v2f;
typedef __attribute__((ext_vector_type(8))) float v8f;

#define IN_DIM 128
#define HID 32

// ---------------- degree / normalization ----------------

__global__ void init_deg_kernel(float* __restrict__ deg, int n) {
  int i = blockIdx.x * blockDim.x + threadIdx.x;
  if (i < n) deg[i] = 1.0f;  // self-loop contributes 1 to every node's degree
}

__global__ void degree_edge_kernel(const long long* __restrict__ col,
                                   float* __restrict__ deg, int E) {
  int e = blockIdx.x * blockDim.x + threadIdx.x;
  if (e < E) unsafeAtomicAdd(&deg[(int)col[e]], 1.0f);
}

__global__ void rsqrt_kernel(float* __restrict__ deg, int n) {
  int i = blockIdx.x * blockDim.x + threadIdx.x;
  if (i < n) deg[i] = rsqrtf(deg[i]);  // deg >= 1 always (self-loop)
}

// ---------------- GEMM via V_WMMA_F32_16X16X4_F32 ----------------
// out[nrows x 32] = (RELU ? relu(in) : in)[nrows x K] @ W[K x 32]
// One wave computes TWO 16-row M-tiles (32 rows) x full 32 output cols
// -> 4 v8f accumulators, each B fragment feeds 2 independent WMMA chains.
//
// Fragments (lane l, vgpr v):
//   A: (m = l&15, k = 2*(l>>4)+v)  -> contiguous float2 from the input row
//   B: (n = l&15, k = 2*(l>>4)+v)  -> single ds_load_b64 from interleaved LDS
//   C: vgpr r, lanes 0-15 -> M=r, N=lane; lanes 16-31 -> M=r+8, N=lane-16
//
// LDS W layout interleaves k-pairs so both B components are adjacent:
//   sW[(k>>1)*64 + 2*n + (k&1)] = W[k][n]
template <int K, bool RELU>
__global__ void gemm_wmma_kernel(const float* __restrict__ in,
                                 const float* __restrict__ W,
                                 float* __restrict__ out, int nrows) {
  __shared__ float sW[K * HID];
  for (int idx = threadIdx.x; idx < K * HID; idx += blockDim.x) {
    int k = idx >> 5;   // HID == 32
    int n = idx & 31;
    sW[(k >> 1) * 64 + 2 * n + (k & 1)] = W[idx];
  }
  __syncthreads();

  int wave = threadIdx.x >> 5;
  int lane = threadIdx.x & 31;
  int rowBase = (blockIdx.x * 4 + wave) * 32;
  if (rowBase >= nrows) return;  // wave-uniform: EXEC stays all-ones for WMMA

  int m  = lane & 15;   // A row within tile / B-C column within tile
  int kh = lane >> 4;   // K-half select

  v8f c00 = {}, c01 = {}, c10 = {}, c11 = {};
  const float* arow0 = in + (size_t)(rowBase + m) * K + 2 * kh;
  const float* arow1 = arow0 + (size_t)16 * K;

  for (int k0 = 0; k0 < K; k0 += 4) {
    int p = (k0 >> 1) + kh;  // k-pair index for this lane
    v2f b0 = *(const v2f*)&sW[p * 64 + 2 * m];         // cols 0..15
    v2f b1 = *(const v2f*)&sW[p * 64 + 2 * (m + 16)];  // cols 16..31

    v2f a0 = *(const v2f*)(arow0 + k0);
    v2f a1 = *(const v2f*)(arow1 + k0);
    if constexpr (RELU) {
      a0.x = fmaxf(a0.x, 0.0f); a0.y = fmaxf(a0.y, 0.0f);
      a1.x = fmaxf(a1.x, 0.0f); a1.y = fmaxf(a1.y, 0.0f);
    }

    c00 = __builtin_amdgcn_wmma_f32_16x16x4_f32(false, a0, false, b0,
                                                (short)0, c00, false, false);
    c01 = __builtin_amdgcn_wmma_f32_16x16x4_f32(false, a0, false, b1,
                                                (short)0, c01, false, false);
    c10 = __builtin_amdgcn_wmma_f32_16x16x4_f32(false, a1, false, b0,
                                                (short)0, c10, false, false);
    c11 = __builtin_amdgcn_wmma_f32_16x16x4_f32(false, a1, false, b1,
                                                (short)0, c11, false, false);
  }

  float* orow0 = out + (size_t)(rowBase + 8 * kh) * HID;
  float* orow1 = orow0 + 16 * HID;
#pragma unroll
  for (int r = 0; r < 8; ++r) {
    orow0[r * HID + m]      = c00[r];
    orow0[r * HID + 16 + m] = c01[r];
    orow1[r * HID + m]      = c10[r];
    orow1[r * HID + 16 + m] = c11[r];
  }
}

// ---------------- aggregation ----------------
// agg[i][c] = bias[c] + h[i][c]*dinv[i]^2   (bias + self-loop term folded in)
__global__ void agg_init_kernel(const float* __restrict__ h,
                                const float* __restrict__ dinv,
                                const float* __restrict__ bias,
                                float* __restrict__ agg, int n) {
  int tid = blockIdx.x * blockDim.x + threadIdx.x;
  if (tid >= n * HID) return;
  int i = tid >> 5, c = tid & 31;
  float d = dinv[i];
  agg[tid] = bias[c] + h[tid] * d * d;
}

// one thread per (edge, channel): coalesced h reads, native f32 atomics.
// H (12.8 MB) fits in the 192 MB L2, so gathers/atomics stay on-chip.
__global__ void edge_scatter32_kernel(const float* __restrict__ h,
                                      const long long* __restrict__ row,
                                      const long long* __restrict__ col,
                                      const float* __restrict__ dinv,
                                      float* __restrict__ agg, int E) {
  long long tid = (long long)blockIdx.x * blockDim.x + threadIdx.x;
  if (tid >= (long long)E * HID) return;
  int e = (int)(tid >> 5), c = (int)(tid & 31);
  int r = (int)row[e], t = (int)col[e];
  float w = dinv[r] * dinv[t];
  unsafeAtomicAdd(&agg[(size_t)t * HID + c], h[(size_t)r * HID + c] * w);
}

// ---------------- layer 3 (HID -> 1) ----------------
__global__ void h3_kernel(const float* __restrict__ agg,
                          const float* __restrict__ W3,
                          float* __restrict__ h3, int n) {
  int i = blockIdx.x * blockDim.x + threadIdx.x;
  if (i >= n) return;
  float s = 0.0f;
#pragma unroll
  for (int c = 0; c < HID; ++c)
    s += fmaxf(agg[(size_t)i * HID + c], 0.0f) * W3[c];  // relu(AGG2) @ W3
  h3[i] = s;
}

__global__ void out_init_kernel(const float* __restrict__ h3,
                                const float* __restrict__ dinv,
                                const float* __restrict__ b3,
                                float* __restrict__ out, int n) {
  int i = blockIdx.x * blockDim.x + threadIdx.x;
  if (i >= n) return;
  float d = dinv[i];
  out[i] = b3[0] + h3[i] * d * d;
}

__global__ void edge_scatter1_kernel(const float* __restrict__ h3,
                                     const long long* __restrict__ row,
                                     const long long* __restrict__ col,
                                     const float* __restrict__ dinv,
                                     float* __restrict__ out, int E) {
  int e = blockIdx.x * blockDim.x + threadIdx.x;
  if (e >= E) return;
  int r = (int)row[e], t = (int)col[e];
  unsafeAtomicAdd(&out[t], h3[r] * dinv[r] * dinv[t]);
}

// ---------------- driver ----------------

extern "C" void kernel_launch(void* const* d_in, const int* in_sizes, int n_in,
                              void* d_out, int out_size, void* d_ws, size_t ws_size,
                              hipStream_t stream) {
  const float*     x   = (const float*)d_in[0];
  const long long* ei  = (const long long*)d_in[1];
  const float*     W1  = (const float*)d_in[2];
  const float*     b1  = (const float*)d_in[3];
  const float*     W2  = (const float*)d_in[4];
  const float*     b2  = (const float*)d_in[5];
  const float*     W3  = (const float*)d_in[6];
  const float*     b3  = (const float*)d_in[7];
  float*           out = (float*)d_out;

  const int N = in_sizes[0] / IN_DIM;
  const int E = in_sizes[1] / 2;
  const long long* row = ei;       // message source (x_j)
  const long long* col = ei + E;   // aggregation target

  float* dinv = (float*)d_ws;
  float* bufA = dinv + N;                    // transformed features h = in @ W
  float* bufB = bufA + (size_t)N * HID;      // aggregated features (pre-relu)
  float* h3   = bufB + (size_t)N * HID;      // layer-3 scalar features

  const int nBlk = (N + 255) / 256;
  const int eBlk = (E + 255) / 256;
  const int gBlk = (N + 127) / 128;          // 4 waves x 32 rows per block
  const unsigned ecBlk = (unsigned)(((long long)E * HID + 255) / 256);
  const int ncBlk = (N * HID + 255) / 256;

  // symmetric deg^-1/2 normalization (self-loops included)
  init_deg_kernel<<<nBlk, 256, 0, stream>>>(dinv, N);
  degree_edge_kernel<<<eBlk, 256, 0, stream>>>(col, dinv, E);
  rsqrt_kernel<<<nBlk, 256, 0, stream>>>(dinv, N);

  // layer 1: H1 = X @ W1 ; AGG1 = scatter(H1) + b1
  gemm_wmma_kernel<IN_DIM, false><<<gBlk, 128, 0, stream>>>(x, W1, bufA, N);
  agg_init_kernel<<<ncBlk, 256, 0, stream>>>(bufA, dinv, b1, bufB, N);
  edge_scatter32_kernel<<<ecBlk, 256, 0, stream>>>(bufA, row, col, dinv, bufB, E);

  // layer 2: H2 = relu(AGG1) @ W2 ; AGG2 = scatter(H2) + b2
  gemm_wmma_kernel<HID, true><<<gBlk, 128, 0, stream>>>(bufB, W2, bufA, N);
  agg_init_kernel<<<ncBlk, 256, 0, stream>>>(bufA, dinv, b2, bufB, N);
  edge_scatter32_kernel<<<ecBlk, 256, 0, stream>>>(bufA, row, col, dinv, bufB, E);

  // layer 3: h3 = relu(AGG2) @ W3 ; out = scatter(h3) + b3
  h3_kernel<<<nBlk, 256, 0, stream>>>(bufB, W3, h3, N);
  out_init_kernel<<<nBlk, 256, 0, stream>>>(h3, dinv, b3, out, N);
  edge_scatter1_kernel<<<eBlk, 256, 0, stream>>>(h3, row, col, dinv, out, E);
}